// Model_11879879543848
// MI455X (gfx1250) — compile-verified
//
#include <hip/hip_runtime.h>
#include <math.h>

// ---------------------------------------------------------------------------
// ANI-AEV mean collapsed to a scalar:
//   S_rad = sum_{i!=j} fc(d,RCR) * 0.25 * sum_r exp(-ETA_R (d-SHF_R[r])^2)
//   S_ang = sum_i sum_{a<b in nbrs(i)} 2 fc(ra)fc(rb) * (sum_z f1) * (sum_s f2)
//   out   = (S_rad + S_ang) / (N * 1904)
// d^2 = |xi|^2+|xj|^2-2 xi.xj via V_WMMA_F32_16X16X4_F32; positions staged
// into LDS with the Tensor Data Mover (TENSOR_LOAD_TO_LDS + s_wait_tensorcnt).
// ---------------------------------------------------------------------------

#define N_ATOMS 2000
#define NTILE   125                 // 2000 / 16, exact
#define NWAVES  8                   // waves per row-panel workgroup
#define NRADPART (NTILE * NWAVES)   // 1000 per-wave partials
#define K_NB    24
#define MAXNB   32
#define RCR_F   5.1f
#define RCA_F   3.5f
#define ETA_R_F 19.7f
#define ZETA_F  14.1f
#define ETA_A_F 12.5f
#define PI_F    3.14159265358979323846f

typedef float v2f __attribute__((ext_vector_type(2)));
typedef float v8f __attribute__((ext_vector_type(8)));
typedef unsigned int v4u __attribute__((ext_vector_type(4)));
typedef int v8i __attribute__((ext_vector_type(8)));
typedef int v4i __attribute__((ext_vector_type(4)));

// ---------------------------------------------------------------------------
// Kernel 0: zero per-atom neighbor counters (ws is poisoned 0xAA by harness).
// ---------------------------------------------------------------------------
__global__ void aev_init(int* __restrict__ cnt) {
    const int i = blockIdx.x * blockDim.x + threadIdx.x;
    if (i < N_ATOMS) cnt[i] = 0;
}

// ---------------------------------------------------------------------------
// Kernel 1: one workgroup (8 waves) per 16-row panel of the NxN distance map.
//  - TDM stages the full positions array (2000x3 f32 = 24 KB) into LDS once.
//  - Each wave sweeps column tiles tj = wave, wave+8, ...: Gram tile via
//    V_WMMA_F32_16X16X4_F32 with a loop-invariant A fragment (unroll x2).
//  - Radial partial per wave (deterministic shuffle tree), neighbor append
//    for d <= RCA.
// A fragment (ISA 7.12.2, 32-bit A 16x4): lanes 0-15 hold M=lane, K={0,1};
// lanes 16-31 hold M=lane-16, K={2,3}.  B mirrors it (4x16).  C/D: lane<16 ->
// c[v]=D[M=v][N=lane]; lane>=16 -> M=v+8, N=lane-16.
// ---------------------------------------------------------------------------
__global__ void __launch_bounds__(256) aev_radial_wmma(
    const float* __restrict__ pos,
    float* __restrict__ radPart,
    int* __restrict__ cnt,
    float4* __restrict__ nbr)
{
    __shared__ float spos[3 * N_ATOMS];          // 24,000 B

    // ---- Tensor Data Mover: DMA positions -> LDS (one issue per WG) ----
    if (threadIdx.x == 0) {
        const unsigned lds_off = (unsigned)(unsigned long long)(const void*)spos;
        const unsigned long long ga = (unsigned long long)pos;
        v4u g0;
        g0[0] = 1u;                                   // count=1, no gather
        g0[1] = lds_off;                              // D#.lds_addr (bytes)
        g0[2] = (unsigned)(ga & 0xffffffffull);       // global_addr[31:0]
        g0[3] = (unsigned)((ga >> 32) & 0x1ffffffull) // global_addr[56:32]
              | (2u << 30);                           // type = 2 ("image")
        v8i g1;
        g1[0] = 2 << 16;                              // data_size = 4 bytes
        g1[1] = (int)((6000u & 0xffffu) << 16);       // tensor_dim0[15:0] @ bits 63:48
        g1[2] = (int)((6000u >> 16) | (1u << 16));    // tensor_dim0[31:16] | tensor_dim1=1
        g1[3] = (int)(6000u << 16);                   // tile_dim0 = 6000 elements
        g1[4] = 1;                                    // tile_dim1 = 1, tile_dim2 = 0
        g1[5] = 6000;                                 // tensor_dim0_stride[31:0]
        g1[6] = (int)((6000u & 0xffffu) << 16);       // dim0_stride[47:32]=0 | dim1_stride[15:0]
        g1[7] = 0;                                    // dim1_stride[47:16]
        v4i g2 = {};                                  // group 2 unused (<=2D tensor)
        v4i g3 = {};                                  // group 3 unused
        v8i g4 = {};                                  // extra descriptor group (clang-23 6-arg form)
        __builtin_amdgcn_tensor_load_to_lds(g0, g1, g2, g3, g4, 0);
        __builtin_amdgcn_s_wait_tensorcnt(0);         // s_wait_tensorcnt 0
    }
    __syncthreads();

    const int  row0 = blockIdx.x * 16;
    const int  wave = threadIdx.x >> 5;
    const int  lane = threadIdx.x & 31;
    const int  g    = lane & 15;
    const bool hi   = lane >= 16;

    // Loop-invariant A fragment (this panel's 16 rows), from LDS.
    const float rx = spos[3 * (row0 + g) + 0];
    const float ry = spos[3 * (row0 + g) + 1];
    const float rz = spos[3 * (row0 + g) + 2];
    v2f a;
    a[0] = hi ? rz : rx;
    a[1] = hi ? 0.0f : ry;

    // Per-v row data for this lane's half of the C tile.
    const int rowBase = row0 + (hi ? 8 : 0);
    float rpx[8], rpy[8], rpz[8], rn[8];
#pragma unroll
    for (int v = 0; v < 8; ++v) {
        rpx[v] = spos[3 * (rowBase + v) + 0];
        rpy[v] = spos[3 * (rowBase + v) + 1];
        rpz[v] = spos[3 * (rowBase + v) + 2];
        rn[v]  = rpx[v] * rpx[v] + rpy[v] * rpy[v] + rpz[v] * rpz[v];
    }

    float acc = 0.0f;
#pragma unroll 2
    for (int tj = wave; tj < NTILE; tj += NWAVES) {
        const int   j  = tj * 16 + g;
        const float cx = spos[3 * j + 0];
        const float cy = spos[3 * j + 1];
        const float cz = spos[3 * j + 2];
        v2f b;
        b[0] = hi ? cz : cx;
        b[1] = hi ? 0.0f : cy;

        v8f c = {};
        c = __builtin_amdgcn_wmma_f32_16x16x4_f32(
                false, a, false, b, (short)0, c, false, false);

        const float nj = cx * cx + cy * cy + cz * cz;
#pragma unroll
        for (int v = 0; v < 8; ++v) {
            const int i = rowBase + v;
            if (i == j) continue;                     // diagonal is masked out
            float d2 = rn[v] + nj - 2.0f * c[v];
            d2 = fmaxf(d2, 0.0f);
            const float d = __builtin_amdgcn_sqrtf(d2);   // single v_sqrt_f32

            if (d <= RCR_F) {                         // fc == 0 beyond RCR
                float s = 0.0f;
#pragma unroll
                for (int r = 0; r < 16; ++r) {
                    const float t = d - (0.8f + 0.26875f * (float)r);
                    s += __expf(-ETA_R_F * t * t);    // v_exp_f32
                }
                const float fc = 0.5f * __cosf(PI_F * d * (1.0f / RCR_F)) + 0.5f;
                acc += 0.25f * s * fc;
            }
            if (d <= RCA_F) {                         // angular neighbor candidate
                const int p = atomicAdd(&cnt[i], 1);
                if (p < MAXNB)
                    nbr[i * MAXNB + p] =
                        make_float4(cx - rpx[v], cy - rpy[v], cz - rpz[v], d);
            }
        }
    }

    // Deterministic wave32 tree reduction, one partial per wave.
#pragma unroll
    for (int off = 16; off > 0; off >>= 1)
        acc += __shfl_xor(acc, off, 32);
    if (lane == 0) radPart[blockIdx.x * NWAVES + wave] = acc;
}

// ---------------------------------------------------------------------------
// Kernel 2: one block per atom.  Sort its neighbor list by distance
// (deterministic + emulates top-K truncation), precompute fc per neighbor,
// then loop unordered pairs.  acos eliminated: cos(th-s)=ca*cos s + sa*sin s.
// ---------------------------------------------------------------------------
__global__ void __launch_bounds__(128) aev_angular(
    const int* __restrict__ cnt,
    const float4* __restrict__ nbr,
    float* __restrict__ angPart)
{
    __shared__ float4 sn[MAXNB];
    __shared__ float  sfc[MAXNB];
    __shared__ float  red[128];

    const int atom = blockIdx.x;
    const int tid  = threadIdx.x;

    __builtin_prefetch(&nbr[atom * MAXNB], 0, 1);     // global_prefetch_b8

    int m = cnt[atom];
    m = (m > MAXNB) ? MAXNB : m;
    if (tid < m) sn[tid] = nbr[atom * MAXNB + tid];
    __syncthreads();

    if (tid == 0 && m > 1) {                          // selection sort by d
        for (int s = 0; s < m - 1; ++s) {
            int best = s;
            for (int t = s + 1; t < m; ++t)
                if (sn[t].w < sn[best].w) best = t;
            if (best != s) { float4 tmp = sn[s]; sn[s] = sn[best]; sn[best] = tmp; }
        }
    }
    __syncthreads();

    if (m > K_NB) m = K_NB;                           // reference keeps 24 nearest
    if (tid < m) sfc[tid] = 0.5f * __cosf(PI_F * sn[tid].w * (1.0f / RCA_F)) + 0.5f;
    __syncthreads();

    // cos/sin of SHF_Z[z] = (2z+1)*pi/8, compile-time constants.
    const float CZ[8] = {  0.92387953f,  0.38268343f, -0.38268343f, -0.92387953f,
                          -0.92387953f, -0.38268343f,  0.38268343f,  0.92387953f };
    const float SZ[8] = {  0.38268343f,  0.92387953f,  0.92387953f,  0.38268343f,
                          -0.38268343f, -0.92387953f, -0.92387953f, -0.38268343f };

    const int P = m * (m - 1) / 2;
    float acc = 0.0f;
    for (int p = tid; p < P; p += blockDim.x) {
        int bb = 1, off = 0;                          // flat pair -> (aa < bb)
        while (off + bb <= p) { off += bb; ++bb; }
        const int aa = p - off;

        const float4 A = sn[aa];
        const float4 B = sn[bb];
        const float  ra = A.w, rb = B.w;
        const float  dotv = A.x * B.x + A.y * B.y + A.z * B.z;
        float ca = __fdividef(0.95f * dotv, ra * rb); // v_rcp_f32 + mul
        ca = fminf(fmaxf(ca, -1.0f), 1.0f);
        const float sa = __builtin_amdgcn_sqrtf(fmaxf(1.0f - ca * ca, 0.0f)); // sin >= 0

        float s1 = 0.0f;
#pragma unroll
        for (int z = 0; z < 8; ++z) {
            float f = 0.5f + 0.5f * (ca * CZ[z] + sa * SZ[z]); // (1+cos(th-shz))/2
            f = fmaxf(f, 0.0f);
            s1 += __powf(f, ZETA_F);                  // v_log_f32 + v_exp_f32
        }
        const float rm = 0.5f * (ra + rb);
        float s2 = 0.0f;
#pragma unroll
        for (int s = 0; s < 8; ++s) {
            const float t = rm - (0.8f + 0.3375f * (float)s);
            s2 += __expf(-ETA_A_F * t * t);
        }
        acc += 2.0f * sfc[aa] * sfc[bb] * s1 * s2;
    }

    red[tid] = acc;
    __syncthreads();
    for (int off = 64; off > 0; off >>= 1) {
        if (tid < off) red[tid] += red[tid + off];
        __syncthreads();
    }
    if (tid == 0) angPart[atom] = red[0];
}

// ---------------------------------------------------------------------------
// Kernel 3: fixed-order final reduction -> mean.
// ---------------------------------------------------------------------------
__global__ void __launch_bounds__(256) aev_reduce(
    const float* __restrict__ radPart,
    const float* __restrict__ angPart,
    float* __restrict__ out)
{
    __shared__ float red[256];
    const int tid = threadIdx.x;
    float acc = 0.0f;
    for (int i = tid; i < NRADPART; i += 256) acc += radPart[i];
    for (int i = tid; i < N_ATOMS;  i += 256) acc += angPart[i];
    red[tid] = acc;
    __syncthreads();
    for (int off = 128; off > 0; off >>= 1) {
        if (tid < off) red[tid] += red[tid + off];
        __syncthreads();
    }
    if (tid == 0) out[0] = red[0] * (1.0f / ((float)N_ATOMS * 1904.0f));
}

// ---------------------------------------------------------------------------
extern "C" void kernel_launch(void* const* d_in, const int* in_sizes, int n_in,
                              void* d_out, int out_size, void* d_ws, size_t ws_size,
                              hipStream_t stream) {
    (void)in_sizes; (void)n_in; (void)out_size; (void)ws_size;
    // d_in[0] = species (int32, N) -- mathematically irrelevant to the mean
    // d_in[1] = positions (float32, N*3)
    const float* pos = (const float*)d_in[1];

    char* ws = (char*)d_ws;
    int*    cnt     = (int*)ws;                       // 2000 * 4      =   8,000 B
    float4* nbr     = (float4*)(ws + 8192);           // 2000*32*16    = 1,024,000 B
    float*  radPart = (float*)(ws + 1032192);         // 1000 * 4      =   4,000 B
    float*  angPart = (float*)(ws + 1036192);         // 2000 * 4      =   8,000 B

    aev_init<<<(N_ATOMS + 255) / 256, 256, 0, stream>>>(cnt);
    aev_radial_wmma<<<NTILE, 256, 0, stream>>>(pos, radPart, cnt, nbr);
    aev_angular<<<N_ATOMS, 128, 0, stream>>>(cnt, nbr, angPart);
    aev_reduce<<<1, 256, 0, stream>>>(radPart, angPart, (float*)d_out);
}